// ConditionedBlockAttentionPairBias_59115929862886
// MI455X (gfx1250) — compile-verified
//
#include <hip/hip_runtime.h>
#include <math.h>

// f32 WMMA fragment types: A/B 16x4 (64 elems / 32 lanes = 2 VGPRs), C/D 16x16 (8 VGPRs)
typedef __attribute__((ext_vector_type(2))) float v2f;
typedef __attribute__((ext_vector_type(8))) float v8f;
typedef __attribute__((ext_vector_type(4))) int v4i;

#define AS1 __attribute__((address_space(1)))
#define AS3 __attribute__((address_space(3)))

#define EPSV 1e-5f
#define INFV 100000.0f

// -------- gfx1250 async global->LDS path (ASYNCcnt), with safe fallback ----
#if defined(__has_builtin)
#if __has_builtin(__builtin_amdgcn_global_load_async_to_lds_b128) && \
    __has_builtin(__builtin_amdgcn_s_wait_asynccnt)
#define USE_ASYNC_LDS 1
#endif
#endif
#ifndef USE_ASYNC_LDS
#define USE_ASYNC_LDS 0
#endif

__device__ __forceinline__ void cp16_async(const float* g, float* l) {
#if USE_ASYNC_LDS
  __builtin_amdgcn_global_load_async_to_lds_b128((AS1 v4i*)g, (AS3 v4i*)l, 0, 0);
#else
  *(float4*)l = *(const float4*)g;
#endif
}
__device__ __forceinline__ void cp_wait_barrier() {
#if USE_ASYNC_LDS
  __builtin_amdgcn_s_wait_asynccnt(0);
#endif
  __syncthreads();
}

__device__ __forceinline__ float waveSum(float v) {
#pragma unroll
  for (int off = 16; off > 0; off >>= 1) v += __shfl_xor(v, off, 32);
  return v;
}
__device__ __forceinline__ float waveMax(float v) {
#pragma unroll
  for (int off = 16; off > 0; off >>= 1) v = fmaxf(v, __shfl_xor(v, off, 32));
  return v;
}
// block-wide sum, blockDim multiple of 32 (<=1024); red holds >= nwaves floats
__device__ __forceinline__ float blockSum(float v, float* red) {
  const int lane = threadIdx.x & 31;
  const int wave = threadIdx.x >> 5;
  const int nw = blockDim.x >> 5;
  v = waveSum(v);
  if (lane == 0) red[wave] = v;
  __syncthreads();
  float t = (lane < nw) ? red[lane] : 0.0f;
  t = waveSum(t);
  __syncthreads();
  return t;
}
__device__ __forceinline__ float sigmoidf(float x) { return 1.0f / (1.0f + expf(-x)); }

// ---------------------------------------------------------------------------
// Kernel 1: cond-side projections (tiny: 64 rows)
// ---------------------------------------------------------------------------
__global__ __launch_bounds__(256) void cond_kernel(
    const float* __restrict__ cond, const float* __restrict__ w_ln_cond,
    const float* __restrict__ w_cond, const float* __restrict__ b_cond,
    const float* __restrict__ w_cond_nb, const float* __restrict__ w_cg,
    const float* __restrict__ b_cg, float* __restrict__ proj1,
    float* __restrict__ proj2, float* __restrict__ proj3) {
  __shared__ float raw[512];
  __shared__ float lnv[512];
  __shared__ float red[8];
  const int mrow = blockIdx.x;
  const float* x = cond + (size_t)mrow * 512;
  float s = 0.0f;
  for (int c = threadIdx.x; c < 512; c += 256) { float v = x[c]; raw[c] = v; s += v; }
  const float mu = blockSum(s, red) * (1.0f / 512.0f);
  float vs = 0.0f;
  for (int c = threadIdx.x; c < 512; c += 256) { float d = raw[c] - mu; vs += d * d; }
  const float var = blockSum(vs, red) * (1.0f / 512.0f);
  const float inv = rsqrtf(var + EPSV);
  for (int c = threadIdx.x; c < 512; c += 256)
    lnv[c] = (raw[c] - mu) * inv * w_ln_cond[c];
  __syncthreads();
  for (int n = threadIdx.x; n < 768; n += 256) {
    const float* w1 = w_cond + (size_t)n * 512;
    const float* w2 = w_cond_nb + (size_t)n * 512;
    const float* w3 = w_cg + (size_t)n * 512;
    float a1 = 0.0f, a2 = 0.0f, a3 = 0.0f;
    for (int c = 0; c < 512; ++c) {
      const float l = lnv[c];
      a1 = fmaf(l, w1[c], a1);
      a2 = fmaf(l, w2[c], a2);
      a3 = fmaf(raw[c], w3[c], a3);
    }
    const size_t o = (size_t)mrow * 768 + n;
    proj1[o] = a1 + b_cond[n];
    proj2[o] = a2;
    proj3[o] = sigmoidf(a3 + b_cg[n]);
  }
}

// ---------------------------------------------------------------------------
// Kernel 2: s = LN(node) * sigmoid(gather(proj1)*m) + gather(proj2)*m
// ---------------------------------------------------------------------------
__global__ __launch_bounds__(256) void s_kernel(
    const float* __restrict__ node, const float* __restrict__ proj1,
    const float* __restrict__ proj2, const int* __restrict__ idx,
    const float* __restrict__ mask, float* __restrict__ s_out) {
  __shared__ float red[8];
  const int i = blockIdx.x;
  const float* x = node + (size_t)i * 768;
  float s = 0.0f;
  for (int c = threadIdx.x; c < 768; c += 256) s += x[c];
  const float mu = blockSum(s, red) * (1.0f / 768.0f);
  float vs = 0.0f;
  for (int c = threadIdx.x; c < 768; c += 256) { float d = x[c] - mu; vs += d * d; }
  const float var = blockSum(vs, red) * (1.0f / 768.0f);
  const float inv = rsqrtf(var + EPSV);
  const int m = idx[i];
  const float mk = mask[i];
  const float* p1 = proj1 + (size_t)m * 768;
  const float* p2 = proj2 + (size_t)m * 768;
  for (int c = threadIdx.x; c < 768; c += 256) {
    const float ln = (x[c] - mu) * inv;
    s_out[(size_t)i * 768 + c] = ln * sigmoidf(p1[c] * mk) + p2[c] * mk;
  }
}

// ---------------------------------------------------------------------------
// Kernel 3: WMMA f32 GEMM  C[MxN] = A[MxK] . W[NxK]^T  with async-LDS staging.
// Block = 256 threads = 8 waves -> 32x64 macro-tile (2x4 grid of 16x16 tiles).
// K consumed in KC=64 chunks, double-buffered in LDS; chunk c+1 streamed with
// GLOBAL_LOAD_ASYNC_TO_LDS_B128 while chunk c feeds V_WMMA_F32_16X16X4_F32.
// Optional epilogue: C *= cgate[idx[row]][col] * mask[row]   (cg2 gather)
// ---------------------------------------------------------------------------
#define KC 64
#define LDT 68  // padded row stride (272B: 16B-aligned, banks spread row*4)
__global__ __launch_bounds__(256) void gemm_atb_kernel(
    const float* __restrict__ A, const float* __restrict__ W,
    float* __restrict__ C, int M, int N, int K,
    const float* __restrict__ cgate, const int* __restrict__ idx,
    const float* __restrict__ mask) {
  __shared__ float Abuf[2][32 * LDT];  // 2 x 8704 B
  __shared__ float Bbuf[2][64 * LDT];  // 2 x 17408 B
  const int tid = threadIdx.x;
  const int lane = tid & 31;
  const int wave = tid >> 5;
  const int tilesN = N >> 6;
  const int bi = blockIdx.x / tilesN;
  const int bj = blockIdx.x - bi * tilesN;
  const int rowA = bi * 32;
  const int rowB = bj * 64;

  // Stage one K-chunk: A 32x64 (512 float4s) + B 64x64 (1024 float4s),
  // 6 x B128 async copies per thread; q ranges are wave-contiguous.
  auto issue = [&](int k0, int buf) {
#pragma unroll
    for (int rr = 0; rr < 6; ++rr) {
      const int q = tid + 256 * rr;
      if (q < 512) {
        const int row = q >> 4, c4 = (q & 15) << 2;
        cp16_async(A + (size_t)(rowA + row) * K + k0 + c4,
                   &Abuf[buf][row * LDT + c4]);
      } else {
        const int qq = q - 512;
        const int row = qq >> 4, c4 = (qq & 15) << 2;
        cp16_async(W + (size_t)(rowB + row) * K + k0 + c4,
                   &Bbuf[buf][row * LDT + c4]);
      }
    }
  };

  const int r = lane & 15;
  const int kh = (lane >> 4) << 1;      // 0 or 2
  const int m0 = (wave >> 2) * 16;      // 0 / 16
  const int n0 = (wave & 3) * 16;       // 0 / 16 / 32 / 48
  const int nch = K / KC;

  v8f acc = {};
  issue(0, 0);
  cp_wait_barrier();
  for (int c = 0; c < nch; ++c) {
    const int b = c & 1;
    if (c + 1 < nch) issue((c + 1) * KC, (c + 1) & 1);
    const float* arow = &Abuf[b][(m0 + r) * LDT + kh];
    const float* brow = &Bbuf[b][(n0 + r) * LDT + kh];
#pragma unroll
    for (int k0 = 0; k0 < KC; k0 += 4) {
      v2f av = {arow[k0], arow[k0 + 1]};
      v2f bv = {brow[k0], brow[k0 + 1]};
      acc = __builtin_amdgcn_wmma_f32_16x16x4_f32(false, av, false, bv,
                                                  (short)0, acc, false, false);
    }
    cp_wait_barrier();  // chunk c+1 landed; all waves done with buf b
  }

  const int n = rowB + n0 + r;
  const int mbase = rowA + m0 + (lane >> 4) * 8;
  if (cgate) {
#pragma unroll
    for (int rr = 0; rr < 8; ++rr) {
      const int mi = mbase + rr;
      const float g = cgate[(size_t)idx[mi] * N + n] * mask[mi];
      C[(size_t)mi * N + n] = acc[rr] * g;
    }
  } else {
#pragma unroll
    for (int rr = 0; rr < 8; ++rr)
      C[(size_t)(mbase + rr) * N + n] = acc[rr];
  }
}

// ---------------------------------------------------------------------------
// Kernel 4: bias[h][i][j] = (LN(edge[i][j])*w_ln_z + b_ln_z) . w_bias[h]
// one wave per (i,j) pair; single streaming pass over the 302MB edge tensor.
// ---------------------------------------------------------------------------
__global__ __launch_bounds__(256) void edge_bias_kernel(
    const float* __restrict__ edge, const float* __restrict__ w_ln_z,
    const float* __restrict__ b_ln_z, const float* __restrict__ w_bias,
    float* __restrict__ bias) {
  __shared__ float wb[12 * 128];
  __shared__ float wz[128];
  __shared__ float bz[128];
  for (int t = threadIdx.x; t < 12 * 128; t += 256) wb[t] = w_bias[t];
  for (int t = threadIdx.x; t < 128; t += 256) { wz[t] = w_ln_z[t]; bz[t] = b_ln_z[t]; }
  __syncthreads();
  const int lane = threadIdx.x & 31;
  const int wave = threadIdx.x >> 5;
  const size_t pair = (size_t)blockIdx.x * 8 + wave;
  const float4 xv = ((const float4*)(edge + pair * 128))[lane];
  const float mu = waveSum(xv.x + xv.y + xv.z + xv.w) * (1.0f / 128.0f);
  const float d0 = xv.x - mu, d1 = xv.y - mu, d2 = xv.z - mu, d3 = xv.w - mu;
  const float var = waveSum(d0 * d0 + d1 * d1 + d2 * d2 + d3 * d3) * (1.0f / 128.0f);
  const float inv = rsqrtf(var + EPSV);
  const int c = lane * 4;
  const float y0 = d0 * inv * wz[c + 0] + bz[c + 0];
  const float y1 = d1 * inv * wz[c + 1] + bz[c + 1];
  const float y2 = d2 * inv * wz[c + 2] + bz[c + 2];
  const float y3 = d3 * inv * wz[c + 3] + bz[c + 3];
  const size_t i = pair / 768, j = pair - i * 768;
#pragma unroll
  for (int hh = 0; hh < 12; ++hh) {
    float p = y0 * wb[hh * 128 + c] + y1 * wb[hh * 128 + c + 1] +
              y2 * wb[hh * 128 + c + 2] + y3 * wb[hh * 128 + c + 3];
    p = waveSum(p);
    if (lane == 0) bias[((size_t)hh * 768 + i) * 768 + j] = p;
  }
}

// ---------------------------------------------------------------------------
// Kernel 5: attention per (head, 16-row tile). 4 waves.
// ---------------------------------------------------------------------------
#define LDSP 776
__global__ __launch_bounds__(128) void attn_kernel(
    const float* __restrict__ q, const float* __restrict__ kv,
    const float* __restrict__ bias, const float* __restrict__ mask,
    float* __restrict__ out) {
  __shared__ float P[16][LDSP];
  const int h = blockIdx.x / 48;
  const int it = blockIdx.x % 48;
  const int lane = threadIdx.x & 31;
  const int wave = threadIdx.x >> 5;  // 0..3
  const int r = lane & 15;
  const int kh = (lane >> 4) << 1;
  const float scale = 0.125f;  // 1/sqrt(64)

  // Phase 1: logits (each wave covers 12 of the 48 j-tiles)
  const float* qrow = q + (size_t)(it * 16 + r) * 768 + h * 64 + kh;
  for (int jt = wave; jt < 48; jt += 4) {
    const float* krow = kv + (size_t)(jt * 16 + r) * 1536 + h * 64 + kh;
    v8f acc = {};
#pragma unroll
    for (int k0 = 0; k0 < 64; k0 += 4) {
      v2f av = {qrow[k0], qrow[k0 + 1]};
      v2f bv = {krow[k0], krow[k0 + 1]};
      acc = __builtin_amdgcn_wmma_f32_16x16x4_f32(false, av, false, bv,
                                                  (short)0, acc, false, false);
    }
    const int j = jt * 16 + r;
    const float mj = mask[j];
    const int mb = (lane >> 4) * 8;
#pragma unroll
    for (int rr = 0; rr < 8; ++rr) {
      const int mi = mb + rr;
      const int gi = it * 16 + mi;
      const float em = mask[gi] * mj;
      P[mi][j] = acc[rr] * scale + bias[((size_t)h * 768 + gi) * 768 + j] +
                 (em - 1.0f) * INFV;
    }
  }
  __syncthreads();

  // Phase 2: softmax, one wave per row (4 rows per wave)
  for (int row = wave; row < 16; row += 4) {
    float mx = -1e30f;
    for (int j = lane; j < 768; j += 32) mx = fmaxf(mx, P[row][j]);
    mx = waveMax(mx);
    float sum = 0.0f;
    for (int j = lane; j < 768; j += 32) {
      const float e = expf(P[row][j] - mx);
      P[row][j] = e;
      sum += e;
    }
    sum = waveSum(sum);
    const float rinv = 1.0f / sum;
    for (int j = lane; j < 768; j += 32) P[row][j] *= rinv;
  }
  __syncthreads();

  // Phase 3: O(16x64) = P(16x768) . V(768x64); one wave per 16-col slice
  {
    const int col = wave * 16 + r;
    const float* vcol = kv + 768 + (size_t)h * 64 + col;  // v[j][col] = vcol[j*1536]
    v8f acc = {};
    for (int k0 = 0; k0 < 768; k0 += 4) {
      const int kk = k0 + kh;
      v2f av = {P[r][kk], P[r][kk + 1]};
      v2f bv = {vcol[(size_t)kk * 1536], vcol[(size_t)(kk + 1) * 1536]};
      acc = __builtin_amdgcn_wmma_f32_16x16x4_f32(false, av, false, bv,
                                                  (short)0, acc, false, false);
    }
    const int n = h * 64 + wave * 16 + r;
    const int mbase = it * 16 + (lane >> 4) * 8;
#pragma unroll
    for (int rr = 0; rr < 8; ++rr)
      out[(size_t)(mbase + rr) * 768 + n] = acc[rr];
  }
}

// ---------------------------------------------------------------------------
// Kernel 6: attn *= sigmoid(gate), in place
// ---------------------------------------------------------------------------
__global__ __launch_bounds__(256) void gate_mul_kernel(
    float* __restrict__ attn, const float* __restrict__ gate, int n) {
  const int t = blockIdx.x * blockDim.x + threadIdx.x;
  if (t < n) attn[t] *= sigmoidf(gate[t]);
}

// ---------------------------------------------------------------------------
extern "C" void kernel_launch(void* const* d_in, const int* in_sizes, int n_in,
                              void* d_out, int out_size, void* d_ws,
                              size_t ws_size, hipStream_t stream) {
  (void)in_sizes; (void)n_in; (void)out_size; (void)ws_size;
  const float* node_embed = (const float*)d_in[0];   // [768,768]
  const float* cond_embed = (const float*)d_in[1];   // [64,512]
  const float* edge_embed = (const float*)d_in[2];   // [768,768,128]
  const float* node_mask  = (const float*)d_in[3];   // [768]
  const int*   idx        = (const int*)d_in[4];     // [768]
  const float* w_ln_cond  = (const float*)d_in[5];
  const float* w_cond     = (const float*)d_in[6];   // [768,512]
  const float* b_cond     = (const float*)d_in[7];
  const float* w_cond_nb  = (const float*)d_in[8];   // [768,512]
  const float* w_q        = (const float*)d_in[9];   // [768,768]
  const float* w_kv       = (const float*)d_in[10];  // [1536,768]
  const float* w_ln_z     = (const float*)d_in[11];  // [128]
  const float* b_ln_z     = (const float*)d_in[12];  // [128]
  const float* w_bias     = (const float*)d_in[13];  // [12,128]
  const float* w_gate     = (const float*)d_in[14];  // [768,768]
  const float* w_cg       = (const float*)d_in[15];  // [768,512]
  const float* b_cg       = (const float*)d_in[16];
  const float* w_out      = (const float*)d_in[17];  // [768,768]
  float* out = (float*)d_out;

  float* ws = (float*)d_ws;
  float* proj1    = ws;                    // 64*768
  float* proj2    = proj1 + 64 * 768;      // 64*768
  float* proj3    = proj2 + 64 * 768;      // 64*768 (cg2, sigmoided)
  float* s_buf    = proj3 + 64 * 768;      // 768*768
  float* q_buf    = s_buf + 768 * 768;     // 768*768
  float* kv_buf   = q_buf + 768 * 768;     // 768*1536
  float* g_buf    = kv_buf + 768 * 1536;   // 768*768
  float* a_buf    = g_buf + 768 * 768;     // 768*768
  float* bias_buf = a_buf + 768 * 768;     // 12*768*768

  cond_kernel<<<64, 256, 0, stream>>>(cond_embed, w_ln_cond, w_cond, b_cond,
                                      w_cond_nb, w_cg, b_cg, proj1, proj2, proj3);
  s_kernel<<<768, 256, 0, stream>>>(node_embed, proj1, proj2, idx, node_mask, s_buf);
  // q / kv / gate projections; grid = (M/32)*(N/64) macro-tiles
  gemm_atb_kernel<<<(768 / 32) * (768 / 64), 256, 0, stream>>>(
      s_buf, w_q, q_buf, 768, 768, 768, nullptr, nullptr, nullptr);
  gemm_atb_kernel<<<(768 / 32) * (1536 / 64), 256, 0, stream>>>(
      s_buf, w_kv, kv_buf, 768, 1536, 768, nullptr, nullptr, nullptr);
  gemm_atb_kernel<<<(768 / 32) * (768 / 64), 256, 0, stream>>>(
      s_buf, w_gate, g_buf, 768, 768, 768, nullptr, nullptr, nullptr);
  // single streaming pass over edge_embed (bandwidth-dominant step)
  edge_bias_kernel<<<(768 * 768) / 8, 256, 0, stream>>>(edge_embed, w_ln_z,
                                                        b_ln_z, w_bias, bias_buf);
  attn_kernel<<<12 * 48, 128, 0, stream>>>(q_buf, kv_buf, bias_buf, node_mask, a_buf);
  gate_mul_kernel<<<(768 * 768 + 255) / 256, 256, 0, stream>>>(a_buf, g_buf, 768 * 768);
  // out = (attn * sig(gate)) @ w_out^T, fused with cg2 gather epilogue
  gemm_atb_kernel<<<(768 / 32) * (768 / 64), 256, 0, stream>>>(
      a_buf, w_out, out, 768, 768, 768, proj3, idx, node_mask);
}